// Merger_34067680592528
// MI455X (gfx1250) — compile-verified
//
#include <hip/hip_runtime.h>

// out = in1*w + in2*(1-w)  ==  fma(w, in1-in2, in2), w broadcast over last
// axis (4096). Pure streaming problem: 768 MB moved, ~128 MFLOP -> HBM-bound
// (floor ~33us @ 23.3 TB/s). Optimized for data movement only:
//   - b128 coalesced loads/stores, non-temporal (inputs 512MB + output 256MB
//     are touched once and exceed the 192MB L2 -> TH=NT keeps L2 clean)
//   - grid-stride chosen as a multiple of FEAT4 so each thread's feature
//     index is loop-invariant: weight float4 loaded ONCE into registers
//     (16KB weight stays L2-resident with regular temporal loads)
//   - 4x manual unroll: 8 independent b128 loads in flight per thread

typedef float v4f __attribute__((ext_vector_type(4)));

#define FEAT   4096
#define FEAT4  (FEAT / 4)   // 1024 float4s per feature row (power of two)
#define BLOCK  256          // 8 wave32s per block

__device__ __forceinline__ v4f lerp4(v4f wv, v4f x, v4f y) {
    v4f r;
    r.x = fmaf(wv.x, x.x - y.x, y.x);
    r.y = fmaf(wv.y, x.y - y.y, y.y);
    r.z = fmaf(wv.z, x.z - y.z, y.z);
    r.w = fmaf(wv.w, x.w - y.w, y.w);
    return r;
}

__global__ __launch_bounds__(BLOCK) void merger_lerp_kernel(
    const v4f* __restrict__ a,    // input_1 as float4
    const v4f* __restrict__ b,    // input_2 as float4
    const v4f* __restrict__ w,    // weight  as float4 (1024 entries)
    v4f* __restrict__ out,
    int n4)                       // total float4 elements (16,777,216)
{
    const int tid    = blockIdx.x * BLOCK + threadIdx.x;
    const int stride = gridDim.x * BLOCK;   // launch guarantees stride % FEAT4 == 0

    // Feature index is loop-invariant because stride is a multiple of FEAT4:
    // load this thread's weight float4 exactly once (cached, L2-resident).
    const v4f wv = w[tid & (FEAT4 - 1)];

    int g = tid;

    // Main loop, unrolled x4: 8 NT b128 loads outstanding before stores.
    for (; g + 3 * stride < n4; g += 4 * stride) {
        const int g0 = g;
        const int g1 = g +     stride;
        const int g2 = g + 2 * stride;
        const int g3 = g + 3 * stride;

        v4f x0 = __builtin_nontemporal_load(&a[g0]);
        v4f x1 = __builtin_nontemporal_load(&a[g1]);
        v4f x2 = __builtin_nontemporal_load(&a[g2]);
        v4f x3 = __builtin_nontemporal_load(&a[g3]);
        v4f y0 = __builtin_nontemporal_load(&b[g0]);
        v4f y1 = __builtin_nontemporal_load(&b[g1]);
        v4f y2 = __builtin_nontemporal_load(&b[g2]);
        v4f y3 = __builtin_nontemporal_load(&b[g3]);

        __builtin_nontemporal_store(lerp4(wv, x0, y0), &out[g0]);
        __builtin_nontemporal_store(lerp4(wv, x1, y1), &out[g1]);
        __builtin_nontemporal_store(lerp4(wv, x2, y2), &out[g2]);
        __builtin_nontemporal_store(lerp4(wv, x3, y3), &out[g3]);
    }

    // Remainder (empty for the 16M-float4 shape with 4096 blocks).
    for (; g < n4; g += stride) {
        v4f x = __builtin_nontemporal_load(&a[g]);
        v4f y = __builtin_nontemporal_load(&b[g]);
        __builtin_nontemporal_store(lerp4(wv, x, y), &out[g]);
    }
}

extern "C" void kernel_launch(void* const* d_in, const int* in_sizes, int n_in,
                              void* d_out, int out_size, void* d_ws, size_t ws_size,
                              hipStream_t stream) {
    const v4f* a   = (const v4f*)d_in[0];   // input_1: (4,4096,4096) f32
    const v4f* b   = (const v4f*)d_in[1];   // input_2: (4,4096,4096) f32
    const v4f* w   = (const v4f*)d_in[2];   // weight:  (4096,)      f32
    v4f*       out = (v4f*)d_out;

    const int n4 = in_sizes[0] / 4;         // 16,777,216 float4s

    // gridDim multiple of 4 => stride (blocks*256 float4s) is a multiple of
    // FEAT4=1024, making each thread's feature index loop-invariant.
    // 4096 blocks x 8 waves = 32K wave32s; 16 float4s per thread, no tail.
    const int blocks = 4096;

    merger_lerp_kernel<<<blocks, BLOCK, 0, stream>>>(a, b, w, out, n4);
}